// GraphMeanAggregationModule_28295244546274
// MI455X (gfx1250) — compile-verified
//
#include <hip/hip_runtime.h>

// Problem constants (match reference).
#define N_NODES  100000
#define D_FEAT   128
#define N_ETYPES 3
#define N_EDGES  1000000
#define D_OUT    (D_FEAT * (1 + N_ETYPES))   // 512 floats per output row

#define WAVE_SIZE        32
#define BLOCK_THREADS    256
#define WAVES_PER_BLOCK  (BLOCK_THREADS / WAVE_SIZE)
#define SCATTER_BLOCKS   4096
#define PIPE_DEPTH       4   // async rows in flight per wave

// CDNA5 async global->LDS path (ASYNCcnt-tracked). Guarded so the kernel
// still compiles (with plain vector loads) if the builtin is absent.
#if __has_builtin(__builtin_amdgcn_global_load_async_to_lds_b128) && \
    __has_builtin(__builtin_amdgcn_s_wait_asynccnt)
#define USE_ASYNC 1
#else
#define USE_ASYNC 0
#endif

// Builtin takes (v4i AS1*, v4i AS3*, int imm_offset, int cpol) per the
// round-1 diagnostic ("int __vector(4) __device__ *").
typedef int v4i __attribute__((vector_size(16)));
typedef __attribute__((address_space(1))) v4i glob_v4i;
typedef __attribute__((address_space(3))) v4i lds_v4i;

// ---------------------------------------------------------------------------
// Kernel 1: out[:, :128] = x ; out[:, 128:512] = 0 ; counts = 0
// One float4 per thread; row = tid>>7 (2048B row = 128 float4).
// ---------------------------------------------------------------------------
__global__ __launch_bounds__(BLOCK_THREADS)
void graph_init_kernel(const float* __restrict__ x,
                       float* __restrict__ out,
                       float* __restrict__ cnt) {
    const int tid = blockIdx.x * blockDim.x + threadIdx.x;
    if (tid < N_ETYPES * N_NODES) cnt[tid] = 0.0f;
    const int total4 = N_NODES * (D_OUT / 4);
    if (tid >= total4) return;
    const int node = tid >> 7;          // D_OUT/4 == 128
    const int q    = tid & 127;
    float4 v;
    if (q < (D_FEAT / 4)) {
        v = ((const float4*)x)[node * (D_FEAT / 4) + q];
    } else {
        v = make_float4(0.f, 0.f, 0.f, 0.f);
    }
    ((float4*)out)[tid] = v;
}

// ---------------------------------------------------------------------------
// Kernel 2: scatter-accumulate. One wave per edge; lane l owns floats
// [4l, 4l+4) of the 128-float feature row (32 lanes x 16B = 512B row).
// Feature rows are staged into LDS with async b128 copies, PIPE_DEPTH deep,
// then accumulated into out with non-returning f32 atomics.
// ---------------------------------------------------------------------------
__global__ __launch_bounds__(BLOCK_THREADS)
void graph_scatter_kernel(const float* __restrict__ x,
                          const int* __restrict__ edges,
                          float* __restrict__ out,
                          float* __restrict__ cnt) {
    const int lane  = threadIdx.x & (WAVE_SIZE - 1);
    const int wave  = threadIdx.x >> 5;
    const int gwave = blockIdx.x * WAVES_PER_BLOCK + wave;
    const int nwaves = SCATTER_BLOCKS * WAVES_PER_BLOCK;
    const int totalE = N_ETYPES * N_EDGES;
    const int chunk  = (totalE + nwaves - 1) / nwaves;
    const int e0 = gwave * chunk;
    const int e1 = min(e0 + chunk, totalE);
    if (e0 >= e1) return;

#if USE_ASYNC
    __shared__ float buf[WAVES_PER_BLOCK][PIPE_DEPTH][D_FEAT];

    // Prologue: fill the pipeline.
    const int npro = min(PIPE_DEPTH, e1 - e0);
    for (int i = 0; i < npro; ++i) {
        const int e   = e0 + i;
        const int t   = e / N_EDGES;
        const int idx = e - t * N_EDGES;
        const int src = edges[t * 2 * N_EDGES + idx];
        const float* gp = x + (size_t)src * D_FEAT + lane * 4;
        __builtin_amdgcn_global_load_async_to_lds_b128(
            (glob_v4i*)gp, (lds_v4i*)&buf[wave][i][lane * 4], 0, 0);
    }

    for (int e = e0; e < e1; ++e) {
        const int slot = (e - e0) & (PIPE_DEPTH - 1);
        const int t    = e / N_EDGES;
        const int idx  = e - t * N_EDGES;
        const int dst  = edges[t * 2 * N_EDGES + N_EDGES + idx];

        // Outstanding asyncs right now = min(PIPE_DEPTH, e1 - e); wait until
        // the oldest (slot's) copy has landed in LDS. Immediate operand only.
        switch (min(PIPE_DEPTH, e1 - e)) {
            case 4:  __builtin_amdgcn_s_wait_asynccnt(3); break;
            case 3:  __builtin_amdgcn_s_wait_asynccnt(2); break;
            case 2:  __builtin_amdgcn_s_wait_asynccnt(1); break;
            default: __builtin_amdgcn_s_wait_asynccnt(0); break;
        }

        float4 v = *(const float4*)&buf[wave][slot][lane * 4];
        // Make sure the ds_load has drained into VGPRs before the slot is
        // refilled by the next async copy.
        asm volatile("s_wait_dscnt 0" ::: "memory");

        // Refill this slot with the row for edge e + PIPE_DEPTH.
        const int en = e + PIPE_DEPTH;
        if (en < e1) {
            const int tn   = en / N_EDGES;
            const int idxn = en - tn * N_EDGES;
            const int srcn = edges[tn * 2 * N_EDGES + idxn];
            const float* gpn = x + (size_t)srcn * D_FEAT + lane * 4;
            __builtin_amdgcn_global_load_async_to_lds_b128(
                (glob_v4i*)gpn, (lds_v4i*)&buf[wave][slot][lane * 4], 0, 0);
        }

        float* ob = out + (size_t)dst * D_OUT + D_FEAT + t * D_FEAT + lane * 4;
        unsafeAtomicAdd(ob + 0, v.x);
        unsafeAtomicAdd(ob + 1, v.y);
        unsafeAtomicAdd(ob + 2, v.z);
        unsafeAtomicAdd(ob + 3, v.w);
        if (lane == 0) unsafeAtomicAdd(&cnt[t * N_NODES + dst], 1.0f);
    }
#else
    for (int e = e0; e < e1; ++e) {
        const int t   = e / N_EDGES;
        const int idx = e - t * N_EDGES;
        const int src = edges[t * 2 * N_EDGES + idx];
        const int dst = edges[t * 2 * N_EDGES + N_EDGES + idx];
        float4 v = *(const float4*)(x + (size_t)src * D_FEAT + lane * 4);
        float* ob = out + (size_t)dst * D_OUT + D_FEAT + t * D_FEAT + lane * 4;
        unsafeAtomicAdd(ob + 0, v.x);
        unsafeAtomicAdd(ob + 1, v.y);
        unsafeAtomicAdd(ob + 2, v.z);
        unsafeAtomicAdd(ob + 3, v.w);
        if (lane == 0) unsafeAtomicAdd(&cnt[t * N_NODES + dst], 1.0f);
    }
#endif
}

// ---------------------------------------------------------------------------
// Kernel 3: mean finalize. out[:, 128+t*128 : 128+(t+1)*128] /= count[t] where
// count > 0 (zero-in-degree rows were zero-initialized and never touched).
// One float4 per thread: 100000 * 3 * 32 threads.
// ---------------------------------------------------------------------------
__global__ __launch_bounds__(BLOCK_THREADS)
void graph_finalize_kernel(float* __restrict__ out,
                           const float* __restrict__ cnt) {
    const int tid = blockIdx.x * blockDim.x + threadIdx.x;
    const int total = N_NODES * (N_ETYPES * D_FEAT / 4);   // N * 96
    if (tid >= total) return;
    const int node = tid / 96;
    const int r    = tid - node * 96;
    const int t    = r >> 5;
    const int q    = r & 31;
    const float c = cnt[t * N_NODES + node];
    if (c > 0.f) {
        float4* p = (float4*)(out + (size_t)node * D_OUT + D_FEAT + t * D_FEAT) + q;
        float4 v = *p;
        v.x /= c; v.y /= c; v.z /= c; v.w /= c;
        *p = v;
    }
}

// ---------------------------------------------------------------------------
extern "C" void kernel_launch(void* const* d_in, const int* in_sizes, int n_in,
                              void* d_out, int out_size, void* d_ws, size_t ws_size,
                              hipStream_t stream) {
    (void)in_sizes; (void)n_in; (void)out_size; (void)ws_size;
    const float* x     = (const float*)d_in[0];
    const int*   edges = (const int*)d_in[1];
    float* out = (float*)d_out;
    float* cnt = (float*)d_ws;   // 3 * N_NODES floats = 1.2 MB of scratch

    const int initTotal = N_NODES * (D_OUT / 4);
    graph_init_kernel<<<(initTotal + BLOCK_THREADS - 1) / BLOCK_THREADS,
                        BLOCK_THREADS, 0, stream>>>(x, out, cnt);

    graph_scatter_kernel<<<SCATTER_BLOCKS, BLOCK_THREADS, 0, stream>>>(
        x, edges, out, cnt);

    const int finTotal = N_NODES * (N_ETYPES * D_FEAT / 4);
    graph_finalize_kernel<<<(finTotal + BLOCK_THREADS - 1) / BLOCK_THREADS,
                            BLOCK_THREADS, 0, stream>>>(out, cnt);
}